// VectorQuantizerEMA_5196910429028
// MI455X (gfx1250) — compile-verified
//
#include <hip/hip_runtime.h>
#include <math.h>
#include <stdint.h>

typedef __attribute__((ext_vector_type(2))) float v2f;
typedef __attribute__((ext_vector_type(8))) float v8f;

#define N_E   4096
#define E_DIM 256
#define N_TOT 16384
#define HW    1024
#define DECAYF 0.99f
#define OMDEC  0.01f
#define BETAF  0.25f
#define EPSF   1e-5f

// ---- d_out layout (floats, tuple concatenated in return order) ----
#define ZQ_ELEMS (16 * 256 * 1024)
#define OFF_LOSS 0
#define OFF_ZQ   1
#define OFF_PERP (OFF_ZQ + ZQ_ELEMS)
#define OFF_IDX  (OFF_PERP + 1)
#define OFF_NEMB (OFF_IDX + N_TOT)
#define OFF_NCS  (OFF_NEMB + N_E * E_DIM)
#define OFF_NEA  (OFF_NCS + N_E)

// ---- d_ws layout (floats) ----
#define WS_COMMIT 0
#define WS_NSUM   1
#define WS_ENT    2
#define WS_NORMS  16
#define WS_COUNTS (WS_NORMS + N_E)

// LDS B-tile geometry: 16 codebook rows x 260-float padded stride (bank-conflict-free
// b64 fragment reads: lane stride = 260 dwords == 4 banks; 16 lanes cover all 64 banks).
#define B_STRIDE      260
#define B_BUF_FLOATS  (16 * B_STRIDE)
#define B_BUF_BYTES   (B_BUF_FLOATS * 4)   // 16640, 16-byte aligned per row (1040 = 65*16)

// k0: per-call init of accumulator regions (d_out is poisoned; we accumulate into it)
__global__ void k0_init(const float* __restrict__ cluster_size,
                        const float* __restrict__ embed_avg,
                        float* __restrict__ out, float* __restrict__ ws) {
  int i = blockIdx.x * 256 + threadIdx.x;
  if (i < N_E * E_DIM) out[OFF_NEA + i] = DECAYF * embed_avg[i];
  if (i < N_E) {
    out[OFF_NCS + i] = DECAYF * cluster_size[i];
    ws[WS_COUNTS + i] = 0.0f;
  }
  if (i < 4) ws[i] = 0.0f;
}

// k1: codebook row norms  ||e_j||^2  (one wave per row)
__global__ void k1_norms(const float* __restrict__ emb, float* __restrict__ ws) {
  int wave = (blockIdx.x * 256 + threadIdx.x) >> 5;
  int lane = threadIdx.x & 31;
  const float* row = emb + (size_t)wave * E_DIM + lane * 8;
  float s = 0.0f;
#pragma unroll
  for (int k = 0; k < 8; ++k) { float v = row[k]; s += v * v; }
#pragma unroll
  for (int off = 16; off >= 1; off >>= 1) s += __shfl_xor(s, off, 32);
  if (lane == 0) ws[WS_NORMS + wave] = s;
}

// k2: fused fp32-WMMA GEMM (zf @ emb^T) + per-row argmin over 4096 codes.
// Block: 256 threads = 8 waves, 128 z-rows per block; wave owns a 16x256 A tile in VGPRs.
// B tiles (16 codes x 256) double-buffered in LDS via CDNA5 async global->LDS DMA
// (global_load_async_to_lds_b128, ASYNCcnt), so the steady state has zero DS stores
// and no staging VGPRs held across the 64-WMMA body.
__global__ void __launch_bounds__(256)
k2_argmin(const float* __restrict__ z, const float* __restrict__ emb,
          const float* __restrict__ ws, float* __restrict__ out) {
  __shared__ float ldsB[2][B_BUF_FLOATS];

  const int tid  = threadIdx.x;
  const int lane = tid & 31;
  const int wv   = tid >> 5;
  const int half = (lane >> 4) & 1;
  const int l15  = lane & 15;

  // ---- load A fragments (WMMA 16x4 fp32 layout): lane<16 K={c0,c0+1}, lane>=16 K={c0+2,c0+3}
  const int rowm = blockIdx.x * 128 + wv * 16 + l15;
  const int bimg = rowm >> 10;            // image index (128-row block never straddles an image)
  const int hw   = rowm & (HW - 1);
  const float* zbase = z + ((size_t)bimg * E_DIM) * HW + hw;

  v2f a[64];
#pragma unroll
  for (int kk = 0; kk < 64; ++kk) {
    int c0 = kk * 4 + half * 2;
    a[kk].x = zbase[(size_t)c0 * HW];
    a[kk].y = zbase[(size_t)(c0 + 1) * HW];
  }

  const float* enorm = ws + WS_NORMS;

  float minv[8];
  int   mini[8];
#pragma unroll
  for (int i = 0; i < 8; ++i) { minv[i] = 3.4e38f; mini[i] = 0; }

  // ---- async global->LDS issue: 4 x b128 per thread covers a 16x256 B chunk (16 KB)
  const uint32_t ldsbase = (uint32_t)(uintptr_t)(&ldsB[0][0]);
  const int jr = tid >> 6;          // this thread's base row contribution pattern:
  const int c4 = tid & 63;          // i = tid + r*256 -> row (i>>6), float4-col (i&63)

  auto issueB = [&](int jc, int buf) {
#pragma unroll
    for (int r = 0; r < 4; ++r) {
      int i  = tid + r * 256;
      int rj = i >> 6;              // 0..15 codebook row within chunk
      int rc = i & 63;              // float4 column 0..63
      uint32_t laddr = ldsbase + (uint32_t)buf * B_BUF_BYTES +
                       (uint32_t)(rj * (B_STRIDE * 4) + rc * 16);
      const float* g = emb + (size_t)(jc * 16 + rj) * E_DIM + rc * 4;
      asm volatile("global_load_async_to_lds_b128 %0, %1, off"
                   :: "v"(laddr), "v"((const void*)g)
                   : "memory");
    }
  };
  (void)jr; (void)c4;

  issueB(0, 0);   // prologue: chunk 0 -> buffer 0

  for (int jc = 0; jc < 256; ++jc) {
    const int cur = jc & 1;

    // own async writes landed in LDS, then barrier so everyone's have
    asm volatile("s_wait_asynccnt 0x0" ::: "memory");
    __syncthreads();

    // kick DMA for the next chunk into the idle buffer (its readers finished
    // before the barrier one iteration ago); latency hides behind 64 WMMAs
    if (jc + 1 < 256) issueB(jc + 1, 1 - cur);

    float en = enorm[jc * 16 + l15];

    v8f acc = {0.f, 0.f, 0.f, 0.f, 0.f, 0.f, 0.f, 0.f};
    const float* bptr = &ldsB[cur][l15 * B_STRIDE + half * 2];
#pragma unroll
    for (int kk = 0; kk < 64; ++kk) {
      v2f b = *(const v2f*)(bptr + kk * 4);
      acc = __builtin_amdgcn_wmma_f32_16x16x4_f32(false, a[kk], false, b,
                                                  (short)0, acc, false, false);
    }

    // running argmin of  d = ||e_j||^2 - 2*S  (||z||^2 constant per row, irrelevant to argmin)
    int j = jc * 16 + l15;
#pragma unroll
    for (int i = 0; i < 8; ++i) {
      float d = fmaf(-2.0f, acc[i], en);
      if (d < minv[i]) { minv[i] = d; mini[i] = j; }  // strict < keeps earliest j (jc ascending)
    }
  }

  // ---- reduce (min, idx) across the 16 lanes of each half; first-index tie-break
#pragma unroll
  for (int i = 0; i < 8; ++i) {
    float v = minv[i];
    int   ix = mini[i];
#pragma unroll
    for (int off = 1; off < 16; off <<= 1) {
      float ov = __shfl_xor(v, off, 32);
      int   oi = __shfl_xor(ix, off, 32);
      if (ov < v || (ov == v && oi < ix)) { v = ov; ix = oi; }
    }
    if (l15 == 0) {
      int m = i + half * 8;  // C VGPR i: lanes0-15 -> M=i, lanes16-31 -> M=i+8
      out[OFF_IDX + blockIdx.x * 128 + wv * 16 + m] = (float)ix;
    }
  }
}

// k3: gather z_q (== straight-through output numerically), commitment-loss sum,
//     counts and embed segment-sums (atomic, directly into EMA output slots).
__global__ void k3_gather(const float* __restrict__ z, const float* __restrict__ emb,
                          float* __restrict__ out, float* __restrict__ ws) {
  __shared__ float red[256];
  int n = blockIdx.x * 256 + threadIdx.x;
  int idxn = (int)out[OFF_IDX + n];
  int bimg = n >> 10;
  int hw = n & (HW - 1);
  const float* erow   = emb + (size_t)idxn * E_DIM;
  const float* zbase  = z + ((size_t)bimg * E_DIM) * HW + hw;
  float* zqbase  = out + OFF_ZQ + ((size_t)bimg * E_DIM) * HW + hw;
  float* neabase = out + OFF_NEA + (size_t)idxn * E_DIM;

  atomicAdd(&out[OFF_NCS + idxn], OMDEC);   // + (1-decay) * enc_sum
  atomicAdd(&ws[WS_COUNTS + idxn], 1.0f);   // raw counts for perplexity

  float csum = 0.0f;
  for (int c = 0; c < E_DIM; ++c) {
    float e  = erow[c];
    float zv = zbase[(size_t)c * HW];
    zqbase[(size_t)c * HW] = e;             // coalesced across threads for fixed c
    float dlt = e - zv;
    csum += dlt * dlt;
    atomicAdd(&neabase[c], OMDEC * zv);     // + (1-decay) * embed_sum
  }

  red[threadIdx.x] = csum;
  __syncthreads();
  for (int s = 128; s > 0; s >>= 1) {
    if (threadIdx.x < s) red[threadIdx.x] += red[threadIdx.x + s];
    __syncthreads();
  }
  if (threadIdx.x == 0) atomicAdd(&ws[WS_COMMIT], red[0]);
}

// k4: n = sum(new_cluster_size);  entropy sum for perplexity
__global__ void k4_stats(const float* __restrict__ out, float* __restrict__ ws) {
  __shared__ float r1[256], r2[256];
  int j = blockIdx.x * 256 + threadIdx.x;
  float cnt = ws[WS_COUNTS + j];
  float p = cnt * (1.0f / (float)N_TOT);
  r1[threadIdx.x] = p * logf(p + 1e-10f);
  r2[threadIdx.x] = out[OFF_NCS + j];
  __syncthreads();
  for (int s = 128; s > 0; s >>= 1) {
    if (threadIdx.x < s) {
      r1[threadIdx.x] += r1[threadIdx.x + s];
      r2[threadIdx.x] += r2[threadIdx.x + s];
    }
    __syncthreads();
  }
  if (threadIdx.x == 0) {
    atomicAdd(&ws[WS_ENT], r1[0]);
    atomicAdd(&ws[WS_NSUM], r2[0]);
  }
}

// k5: new_embedding = new_embed_avg / cluster_smoothed; scalars
__global__ void k5_final(float* __restrict__ out, const float* __restrict__ ws) {
  if (blockIdx.x == N_E) {
    if (threadIdx.x == 0) {
      out[OFF_LOSS] = BETAF * ws[WS_COMMIT] * (1.0f / (float)ZQ_ELEMS);
      out[OFF_PERP] = expf(-ws[WS_ENT]);
    }
    return;
  }
  int j = blockIdx.x;
  float n  = ws[WS_NSUM];
  float cs = out[OFF_NCS + j];
  float sm = (cs + EPSF) / (n + (float)N_E * EPSF) * n;
  int c = threadIdx.x;
  out[OFF_NEMB + (size_t)j * E_DIM + c] = out[OFF_NEA + (size_t)j * E_DIM + c] / sm;
}

extern "C" void kernel_launch(void* const* d_in, const int* in_sizes, int n_in,
                              void* d_out, int out_size, void* d_ws, size_t ws_size,
                              hipStream_t stream) {
  const float* z   = (const float*)d_in[0];  // (16,256,32,32)
  const float* emb = (const float*)d_in[1];  // (4096,256)
  const float* cs  = (const float*)d_in[2];  // (4096,)
  const float* ea  = (const float*)d_in[3];  // (4096,256)
  float* out = (float*)d_out;
  float* ws  = (float*)d_ws;

  k0_init  <<<(N_E * E_DIM + 255) / 256, 256, 0, stream>>>(cs, ea, out, ws);
  k1_norms <<<N_E / 8,       256, 0, stream>>>(emb, ws);
  k2_argmin<<<N_TOT / 128,   256, 0, stream>>>(z, emb, ws, out);
  k3_gather<<<N_TOT / 256,   256, 0, stream>>>(z, emb, out, ws);
  k4_stats <<<N_E / 256,     256, 0, stream>>>(out, ws);
  k5_final <<<N_E + 1,       256, 0, stream>>>(out, ws);
}